// MultiHeadAttention_37323265802898
// MI455X (gfx1250) — compile-verified
//
#include <hip/hip_runtime.h>

// MI455X / gfx1250, wave32. All GEMMs via V_WMMA_F32_16X16X4_F32 (full fp32).

#define B_  2
#define S_  2048
#define E_  768
#define H_  12
#define DK_ 64

typedef __attribute__((ext_vector_type(2))) float v2f;
typedef __attribute__((ext_vector_type(8))) float v8f;

__device__ __forceinline__ v8f wmma_f32(v2f a, v2f b, v8f c) {
  // 8 args: (neg_a, A, neg_b, B, c_mod, C, reuse_a, reuse_b)
  return __builtin_amdgcn_wmma_f32_16x16x4_f32(
      false, a, false, b, (short)0, c, false, false);
}

// ---------------------------------------------------------------------------
// Kernel 1: fused QKV projection. Y = X @ Wk^T + bk for X in {key,query,value},
// output written head-major as [B,H,S,DK].
// One wave computes a 16x32 output tile (2 accumulators, shared A fragment).
// tiles = 3 * (B*S/16) * (E/32) = 3*256*24 = 18432 -> 2304 blocks x 8 waves.
// ---------------------------------------------------------------------------
__global__ void __launch_bounds__(256) proj_qkv(
    const float* __restrict__ key, const float* __restrict__ query,
    const float* __restrict__ value, const float* __restrict__ Wk,
    const float* __restrict__ bk,
    float* __restrict__ qb, float* __restrict__ kb, float* __restrict__ vb)
{
  const int lane = threadIdx.x & 31;
  const int wave = threadIdx.x >> 5;
  const int half = lane >> 4;       // 0: K=0,1   1: K=2,3 (A/B fragment halves)
  const int l16  = lane & 15;

  const int MT = (B_ * S_) / 16;    // 256
  const int NT = E_ / 32;           // 24
  int tile  = blockIdx.x * 8 + wave;
  int which = tile / (MT * NT);     // 0=key,1=query,2=value
  int rem   = tile % (MT * NT);
  int mt = rem / NT, nt = rem % NT;

  const float* X = (which == 0) ? key : (which == 1) ? query : value;
  float*       Y = (which == 0) ? kb  : (which == 1) ? qb    : vb;

  const int arow  = mt * 16 + l16;      // A-matrix row this lane feeds
  const int bcol0 = nt * 32 + l16;      // output cols (rows of Wk)
  const int bcol1 = bcol0 + 16;

  const float* Arow = X  + (size_t)arow  * E_ + half * 2;
  const float* B0   = Wk + (size_t)bcol0 * E_ + half * 2;
  const float* B1   = Wk + (size_t)bcol1 * E_ + half * 2;

  v8f acc0 = {}, acc1 = {};
#pragma unroll 8
  for (int k0 = 0; k0 < E_; k0 += 4) {
    float2 ta = *(const float2*)(Arow + k0);
    float2 t0 = *(const float2*)(B0 + k0);
    float2 t1 = *(const float2*)(B1 + k0);
    v2f a;  a.x  = ta.x; a.y  = ta.y;
    v2f b0; b0.x = t0.x; b0.y = t0.y;
    v2f b1; b1.x = t1.x; b1.y = t1.y;
    acc0 = wmma_f32(a, b0, acc0);
    acc1 = wmma_f32(a, b1, acc1);
  }

  // Whole 16-row tile lies within one batch (S % 16 == 0): hoist the divide.
  const int r0 = mt * 16;
  const int bb = r0 / S_, s0 = r0 % S_;
  const float bias0 = bk[bcol0], bias1 = bk[bcol1];
  const int h0 = bcol0 / DK_, d0 = bcol0 % DK_;
  const int h1 = bcol1 / DK_, d1 = bcol1 % DK_;
  float* Y0 = Y + (((size_t)bb * H_ + h0) * S_ + s0) * DK_ + d0;
  float* Y1 = Y + (((size_t)bb * H_ + h1) * S_ + s0) * DK_ + d1;
#pragma unroll
  for (int g = 0; g < 8; ++g) {
    int m = half ? (g + 8) : g;
    Y0[(size_t)m * DK_] = acc0[g] + bias0;
    Y1[(size_t)m * DK_] = acc1[g] + bias1;
  }
}

// ---------------------------------------------------------------------------
// Kernel 2: fused attention (flash-style, no S x S matrix in HBM).
// One wave handles 16 query rows of one (b,h); loops over 128 key blocks.
// scores = (Q K^T) * sqrt(DK); mask==0 -> 1e-6 (pre-softmax, per reference);
// online softmax; O += P V. waves = 24 * 128 = 3072 -> 384 blocks.
// ---------------------------------------------------------------------------
__global__ void __launch_bounds__(256) attn_fused(
    const float* __restrict__ q, const float* __restrict__ k,
    const float* __restrict__ v, const int* __restrict__ mask,
    float* __restrict__ ctx)
{
  __shared__ float pbuf[8][16][16];   // per-wave P-tile staging (C->A relayout)

  const int lane = threadIdx.x & 31;
  const int wave = threadIdx.x >> 5;
  const int half = lane >> 4;
  const int l16  = lane & 15;

  const int RB = S_ / 16;             // 128 row-blocks per (b,h)
  int wid = blockIdx.x * 8 + wave;
  int bh  = wid / RB;
  int rb  = wid % RB;

  const float* Q = q + (size_t)bh * S_ * DK_;
  const float* K = k + (size_t)bh * S_ * DK_;
  const float* V = v + (size_t)bh * S_ * DK_;

  // Preload Q A-fragments for all 16 K-steps (DK=64 = 16 x K4)
  const int qrow = rb * 16 + l16;
  v2f qa[16];
#pragma unroll
  for (int kk = 0; kk < 16; ++kk) {
    int kd = kk * 4 + half * 2;
    float2 t = *(const float2*)(Q + (size_t)qrow * DK_ + kd);
    qa[kk].x = t.x; qa[kk].y = t.y;
  }

  // Per-g mask row offsets (rows fixed across the whole key loop).
  size_t moff[8];
#pragma unroll
  for (int g = 0; g < 8; ++g) {
    int m = half ? (g + 8) : g;
    moff[g] = (size_t)(rb * 16 + m) * S_ + l16;
  }

  float mrun[8], lrun[8];
  v8f o[4] = {v8f{}, v8f{}, v8f{}, v8f{}};
#pragma unroll
  for (int g = 0; g < 8; ++g) { mrun[g] = -3.0e38f; lrun[g] = 0.0f; }

  for (int t = 0; t < RB; ++t) {
    const int t0 = t * 16;

    // --- scores tile: Q(16x64) @ K_block^T(64x16) ---
    v8f sc = {};
    const float* Krow = K + (size_t)(t0 + l16) * DK_ + half * 2;
#pragma unroll
    for (int kk = 0; kk < 16; ++kk) {
      float2 tb = *(const float2*)(Krow + kk * 4);
      v2f b; b.x = tb.x; b.y = tb.y;
      sc = wmma_f32(qa[kk], b, sc);
    }

    // --- scale by sqrt(DK)=8 and apply mask quirk (masked -> 1e-6) ---
#pragma unroll
    for (int g = 0; g < 8; ++g) {
      float s = sc[g] * 8.0f;
      if (mask[moff[g] + t0] == 0) s = 1e-6f;
      sc[g] = s;
    }

    // --- online softmax update (reductions stay within 16-lane halves) ---
#pragma unroll
    for (int g = 0; g < 8; ++g) {
      float x  = sc[g];
      float mx = x;
#pragma unroll
      for (int off = 1; off < 16; off <<= 1)
        mx = fmaxf(mx, __shfl_xor(mx, off, 32));
      float mnew = fmaxf(mrun[g], mx);
      float p    = __expf(x - mnew);
      float rs   = p;
#pragma unroll
      for (int off = 1; off < 16; off <<= 1)
        rs += __shfl_xor(rs, off, 32);
      float corr = __expf(mrun[g] - mnew);
      lrun[g] = lrun[g] * corr + rs;
      mrun[g] = mnew;
      sc[g]   = p;
      o[0][g] *= corr; o[1][g] *= corr; o[2][g] *= corr; o[3][g] *= corr;
    }

    // --- stage P through LDS: C-layout -> A-layout (same-wave DS, in-order) ---
#pragma unroll
    for (int g = 0; g < 8; ++g) {
      int m = half ? (g + 8) : g;
      pbuf[wave][m][l16] = sc[g];
    }

    // --- O += P(16x16) @ V_block(16x64) ---
#pragma unroll
    for (int kk = 0; kk < 4; ++kk) {
      int kbi = kk * 4 + half * 2;          // key index within block
      v2f a; a.x = pbuf[wave][l16][kbi]; a.y = pbuf[wave][l16][kbi + 1];
      const float* V0 = V + (size_t)(t0 + kbi) * DK_ + l16;
      const float* V1 = V0 + DK_;
#pragma unroll
      for (int nb = 0; nb < 4; ++nb) {
        v2f b; b.x = V0[nb * 16]; b.y = V1[nb * 16];
        o[nb] = wmma_f32(a, b, o[nb]);
      }
    }
  }

  // --- normalize and write context in [B,H,S,DK] ---
#pragma unroll
  for (int g = 0; g < 8; ++g) {
    int m = half ? (g + 8) : g;
    int r = rb * 16 + m;
    float inv = 1.0f / lrun[g];
#pragma unroll
    for (int nb = 0; nb < 4; ++nb)
      ctx[((size_t)bh * S_ + r) * DK_ + nb * 16 + l16] = o[nb][g] * inv;
  }
}

// ---------------------------------------------------------------------------
// Kernel 3: output projection. out[B*S, E] = ctx @ Wo^T + bo, where ctx is
// [B,H,S,DK] and gathered back to [B,S,E] row order in the A-operand load.
// One wave computes a 16x32 tile. tiles = 256*24 = 6144 -> 768 blocks.
// ---------------------------------------------------------------------------
__global__ void __launch_bounds__(256) out_proj(
    const float* __restrict__ ctx, const float* __restrict__ Wo,
    const float* __restrict__ bo, float* __restrict__ out)
{
  const int lane = threadIdx.x & 31;
  const int wave = threadIdx.x >> 5;
  const int half = lane >> 4;
  const int l16  = lane & 15;

  const int NT = E_ / 32;             // 24
  int tile = blockIdx.x * 8 + wave;
  int mt = tile / NT, nt = tile % NT;

  const int arow = mt * 16 + l16;     // row in [0, B*S)
  const int ab   = arow / S_, as = arow % S_;
  const int bcol0 = nt * 32 + l16;
  const int bcol1 = bcol0 + 16;

  // ctx row base for this lane's A row: [B,H,S,DK] gather. (ka, ka+1) always
  // share a head since ka is even and DK is even.
  const float* Abase = ctx + ((size_t)ab * H_ * S_ + as) * DK_;
  const float* B0 = Wo + (size_t)bcol0 * E_ + half * 2;
  const float* B1 = Wo + (size_t)bcol1 * E_ + half * 2;

  v8f acc0 = {}, acc1 = {};
#pragma unroll 8
  for (int k0 = 0; k0 < E_; k0 += 4) {
    int ka = k0 + half * 2;
    int h = ka / DK_, d = ka % DK_;
    float2 ta = *(const float2*)(Abase + (size_t)h * S_ * DK_ + d);
    float2 t0 = *(const float2*)(B0 + k0);
    float2 t1 = *(const float2*)(B1 + k0);
    v2f a;  a.x  = ta.x; a.y  = ta.y;
    v2f b0; b0.x = t0.x; b0.y = t0.y;
    v2f b1; b1.x = t1.x; b1.y = t1.y;
    acc0 = wmma_f32(a, b0, acc0);
    acc1 = wmma_f32(a, b1, acc1);
  }

  const float bias0 = bo[bcol0], bias1 = bo[bcol1];
  float* O0 = out + (size_t)(mt * 16) * E_ + bcol0;
  float* O1 = out + (size_t)(mt * 16) * E_ + bcol1;
#pragma unroll
  for (int g = 0; g < 8; ++g) {
    int m = half ? (g + 8) : g;
    O0[(size_t)m * E_] = acc0[g] + bias0;
    O1[(size_t)m * E_] = acc1[g] + bias1;
  }
}

// ---------------------------------------------------------------------------
extern "C" void kernel_launch(void* const* d_in, const int* in_sizes, int n_in,
                              void* d_out, int out_size, void* d_ws, size_t ws_size,
                              hipStream_t stream) {
  (void)in_sizes; (void)n_in; (void)out_size; (void)ws_size;
  const float* key   = (const float*)d_in[0];
  const float* query = (const float*)d_in[1];
  const float* value = (const float*)d_in[2];
  const int*   mask  = (const int*)d_in[3];
  const float* Wk    = (const float*)d_in[4];
  const float* bk    = (const float*)d_in[5];
  const float* Wo    = (const float*)d_in[6];
  const float* bo    = (const float*)d_in[7];
  float* out = (float*)d_out;

  float* ws = (float*)d_ws;
  const size_t QKV = (size_t)B_ * H_ * S_ * DK_;   // 3,145,728 floats (12 MB)
  float* qb  = ws;
  float* kb  = ws + QKV;
  float* vb  = ws + 2 * QKV;
  float* cb  = ws + 3 * QKV;

  // 3*256*24 tiles / 8 waves per block
  proj_qkv<<<2304, 256, 0, stream>>>(key, query, value, Wk, bk, qb, kb, vb);
  // 24*(2048/16) waves / 8 per block
  attn_fused<<<384, 256, 0, stream>>>(qb, kb, vb, mask, cb);
  // 256*24 tiles / 8 per block
  out_proj<<<768, 256, 0, stream>>>(cb, Wo, bo, out);
}